// GCNFirst_52913997086748
// MI455X (gfx1250) — compile-verified
//
#include <hip/hip_runtime.h>

// CDNA5 / gfx1250. wave32.
typedef __attribute__((ext_vector_type(2))) float v2f;
typedef __attribute__((ext_vector_type(8))) float v8f;

#define THREADS 256  // 8 waves per block on wave32

// Zero the output (n*16 floats) and the degree workspace (n floats).
__global__ void zero_out_and_deg(float* __restrict__ out, float* __restrict__ deg,
                                 long long nOut, long long nDeg) {
  long long i = (long long)blockIdx.x * blockDim.x + threadIdx.x;
  if (i < nOut) out[i] = 0.0f;
  if (i < nDeg) deg[i] = 0.0f;
}

// deg[src] += 1 per edge (exact for integer-valued f32 sums).
__global__ void accum_deg(const int* __restrict__ esrc, float* __restrict__ deg, int E) {
  int i = blockIdx.x * blockDim.x + threadIdx.x;
  if (i < E) {
    int s = esrc[i];             // int offset -> saddr+voffset atomic form
    atomicAdd(deg + s, 1.0f);
  }
}

// deg[i] = 1 / (deg[i]==0 ? 1 : deg[i])  (matches reference where()).
__global__ void invert_deg(float* __restrict__ deg, int n) {
  int i = blockIdx.x * blockDim.x + threadIdx.x;
  if (i < n) {
    float d = deg[i];
    deg[i] = 1.0f / (d == 0.0f ? 1.0f : d);
  }
}

// One wave handles 16 edges per iteration:
//   D(16x16) = diag(inv_deg[src(e)]) * G(16x16 gathered weight rows)
// via 4x V_WMMA_F32_16X16X4_F32 (K=16 split into 4 chunks), then per-lane
// atomic scatter-add of D rows into out[src].
//
// Fast path (full 16-edge group; always taken when E%16==0): branch-free —
// lane L loads src/rel/dst of edge N=L%16 once (coalesced 64B), per-chunk B
// row bases and scatter srcs broadcast via ds_bpermute, all indices kept in
// 32-bit so loads AND atomics use the saddr + voffset + scale_offset form.
__global__ void __launch_bounds__(THREADS)
gcn_scatter_wmma(const float* __restrict__ weights,
                 const int* __restrict__ esrc,
                 const int* __restrict__ erel,
                 const int* __restrict__ edst,
                 const float* __restrict__ inv_deg,
                 float* __restrict__ out,
                 int n, int E) {
  const int lane = threadIdx.x & 31;
  const int N    = lane & 15;   // feature column (B/C/D lane index), also A row M
  const int half = lane >> 4;   // 0: lanes 0-15, 1: lanes 16-31
  const int wavesPerBlock = blockDim.x >> 5;

  const long long nGroups = ((long long)E + 15) >> 4;
  const long long gStride = (long long)gridDim.x * wavesPerBlock;
  long long g = (long long)blockIdx.x * wavesPerBlock + (threadIdx.x >> 5);

  const int relStride = n * 16;  // floats per relation slab (r*n*16 < 2^31 here)

  for (; g < nGroups; g += gStride) {
    const long long ebase = g << 4;

    if (ebase + 16 <= (long long)E) {
      // ---------------- fast path: full group, no per-element guards -------
      const long long eN = ebase + N;           // this lane's "own" edge
      const int   sN   = esrc[eN];              // coalesced 64B per 16 lanes
      const int   rN   = erel[eN];
      const int   dN   = edst[eN];
      const float valM = inv_deg[sN];           // A diagonal value for row N
      const int   rowN = rN * relStride + dN * 16;  // weight-row base (elems)

      v8f acc = {0.f, 0.f, 0.f, 0.f, 0.f, 0.f, 0.f, 0.f};

      #pragma unroll
      for (int c = 0; c < 4; ++c) {
        const int k0 = 4 * c + 2 * half;  // chunk-local K pair for this lane

        // A chunk (16x4): columns k0, k0+1 of the 16x16 diagonal.
        v2f a;
        a.x = (N == k0    ) ? valM : 0.0f;
        a.y = (N == k0 + 1) ? valM : 0.0f;

        // B chunk (4x16): rows = weight rows of edges ebase+k0, ebase+k0+1.
        // Row bases broadcast from the owning lanes via shuffle; the 16 lanes
        // of a half then read 16 consecutive floats -> 64B coalesced.
        const int row0 = __shfl(rowN, k0,     32);
        const int row1 = __shfl(rowN, k0 + 1, 32);
        v2f b;
        b.x = weights[row0 + N];
        b.y = weights[row1 + N];

        // D = A*B + C : exact f32 scale (one nonzero per row of A).
        acc = __builtin_amdgcn_wmma_f32_16x16x4_f32(false, a, false, b,
                                                    (short)0, acc, false, false);
      }

      // C/D layout: lane -> N = lane%16, VGPR v -> M = v + 8*(lane/16).
      // Scatter row M (edge ebase+M); src broadcast via shuffle, 32-bit
      // element offsets so atomics use saddr+voffset+scale_offset, and all 8
      // issue back-to-back with no dependent global loads.
      #pragma unroll
      for (int v = 0; v < 8; ++v) {
        const int m   = v + 8 * half;
        const int s   = __shfl(sN, m, 32);
        const int idx = s * 16 + N;            // < n*16, fits int
        atomicAdd(out + idx, acc[v]);
      }
    } else {
      // ---------------- cold tail path (< 16 edges; never taken if E%16==0)
      #pragma unroll
      for (int v = 0; v < 8; ++v) {
        const int m = v + 8 * half;
        const long long em = ebase + m;
        if (em < E) {
          const int s = esrc[em];
          const int r = erel[em];
          const int d = edst[em];
          const float w = weights[r * relStride + d * 16 + N];
          atomicAdd(out + (s * 16 + N), inv_deg[s] * w);
        }
      }
    }
  }
}

extern "C" void kernel_launch(void* const* d_in, const int* in_sizes, int n_in,
                              void* d_out, int out_size, void* d_ws, size_t ws_size,
                              hipStream_t stream) {
  const float* weights = (const float*)d_in[0];  // (r, n, 16) f32
  const int*   esrc    = (const int*)d_in[1];    // (E,)
  const int*   erel    = (const int*)d_in[2];    // (E,)
  const int*   edst    = (const int*)d_in[3];    // (E,)
  float* out = (float*)d_out;                    // (n, 16) f32
  float* deg = (float*)d_ws;                     // n floats of scratch

  const int E = in_sizes[1];
  const int n = out_size / 16;                   // e == 16

  const long long nOut = (long long)n * 16;
  int blocksZero = (int)((nOut + THREADS - 1) / THREADS);
  zero_out_and_deg<<<blocksZero, THREADS, 0, stream>>>(out, deg, nOut, (long long)n);

  int blocksDeg = (E + THREADS - 1) / THREADS;
  accum_deg<<<blocksDeg, THREADS, 0, stream>>>(esrc, deg, E);

  int blocksInv = (n + THREADS - 1) / THREADS;
  invert_deg<<<blocksInv, THREADS, 0, stream>>>(deg, n);

  const long long nGroups = ((long long)E + 15) / 16;
  const int wavesPerBlock = THREADS / 32;
  long long blocksS = (nGroups + wavesPerBlock - 1) / wavesPerBlock;
  if (blocksS > (1LL << 22)) blocksS = (1LL << 22);  // grid-stride covers the rest
  gcn_scatter_wmma<<<(int)blocksS, THREADS, 0, stream>>>(weights, esrc, erel, edst,
                                                         deg, out, n, E);
}